// GPML_VAE_38397007626881
// MI455X (gfx1250) — compile-verified
//
#include <hip/hip_runtime.h>
#include <cstddef>

// Problem constants (match reference): B=256, T=64, L=32, D=128, S=4

typedef float v2f __attribute__((ext_vector_type(2)));
typedef float v8f __attribute__((ext_vector_type(8)));

// ---------------------------------------------------------------------------
// Kernel A: per-latent prior prep. One 64-thread block per latent l.
//  - Cholesky of K[:,:,l] in LDS -> logdet_K[l]
//  - K^-1 via per-thread forward/back triangular solves (thread j = column j)
//  - store Kinv as [L,T,T] (contiguous tile for the WMMA GEMM) and as a
//    packed lower triangle [L, T*(T+1)/2] for the streaming trace kernel.
// ---------------------------------------------------------------------------
__global__ __launch_bounds__(64) void prior_prep_kernel(
    const float* __restrict__ K, float* __restrict__ KinvL,
    float* __restrict__ KinvP, float* __restrict__ logdetK) {
  __shared__ float Km[64 * 65];  // row stride 65: bank-conflict-free columns
  __shared__ float Yb[64 * 64];  // solve workspace, column j owned by thread j
  __shared__ float red[64];
  const int l = blockIdx.x;
  const int tid = threadIdx.x;

  for (int f = tid; f < 4096; f += 64) {
    int i = f >> 6, j = f & 63;
    Km[i * 65 + j] = K[(size_t)f * 32 + l];  // K layout [T,T,L]
  }
  __syncthreads();

  // Right-looking Cholesky, one row per thread.
  const int r = tid;
  float logacc = 0.0f;
  for (int k = 0; k < 64; ++k) {
    if (r == k) {
      float d = sqrtf(Km[k * 65 + k]);
      Km[k * 65 + k] = d;
      logacc += 2.0f * logf(d);
    }
    __syncthreads();
    float c = 0.0f;
    if (r > k) {
      c = Km[r * 65 + k] / Km[k * 65 + k];
      Km[r * 65 + k] = c;
    }
    __syncthreads();
    if (r > k)
      for (int j = k + 1; j <= r; ++j) Km[r * 65 + j] -= c * Km[j * 65 + k];
    __syncthreads();
  }
  red[tid] = logacc;
  __syncthreads();
  for (int s2 = 32; s2 > 0; s2 >>= 1) {
    if (tid < s2) red[tid] += red[tid + s2];
    __syncthreads();
  }
  if (tid == 0) logdetK[l] = red[0];

  // Kinv column j: solve L y = e_j (forward), then L^T x = y (backward).
  const int j = tid;
  for (int i = 0; i < 64; ++i) {
    float s = (i == j) ? 1.0f : 0.0f;
    for (int p = 0; p < i; ++p) s -= Km[i * 65 + p] * Yb[p * 64 + j];
    Yb[i * 64 + j] = s / Km[i * 65 + i];  // zero for i<j by construction
  }
  for (int i = 63; i >= 0; --i) {
    float s = Yb[i * 64 + j];
    for (int p = i + 1; p < 64; ++p) s -= Km[p * 65 + i] * Yb[p * 64 + j];
    Yb[i * 64 + j] = s / Km[i * 65 + i];
  }
  for (int i = 0; i < 64; ++i) {
    float v = Yb[i * 64 + j];
    KinvL[(size_t)l * 4096 + i * 64 + j] = v;  // [L,T,T]
    if (i >= j) KinvP[(size_t)l * 2080 + ((i * (i + 1)) >> 1) + j] = v;
  }
}

// ---------------------------------------------------------------------------
// Kernel B: the dominant streaming kernel. One 512-thread block per
// (b, 8-latent group). Sigma is symmetric, so only the packed lower triangle
// is ever loaded (halves HBM traffic); 2x float4 per cell covers 8 adjacent
// latents (4x less L2->WGP inflation than float2/2-latent). Fused:
//   trace = sum(diag Kinv.Sig) + 2*sum(lower Kinv.Sig)   (from LDS + KinvP)
//   logdet_S via in-LDS packed Cholesky (row-per-thread, 3-barrier scheme)
// Emits tlds[b,l] = trace - logdet_S.
// ---------------------------------------------------------------------------
__global__ __launch_bounds__(512) void sigma_kernel(
    const float* __restrict__ Sig, const float* __restrict__ KinvP,
    float* __restrict__ tlds) {
  __shared__ float Ap[8 * 2080];  // 8 packed lower-triangular 64x64 blocks
  __shared__ float wred[16];
  const int g = blockIdx.x & 3;
  const int b = blockIdx.x >> 2;
  const int l0 = g * 8;
  const int tid = threadIdx.x;
  const float* Sb = Sig + (size_t)b * 4096 * 32 + l0;  // [B,T,T,L]

  // Load packed lower triangle of the 8 Sigma blocks.
  for (int p = tid; p < 2080; p += 512) {
    int i = (int)((sqrtf(8.0f * (float)p + 1.0f) - 1.0f) * 0.5f);
    while ((i + 1) * (i + 2) / 2 <= p) ++i;
    while (i * (i + 1) / 2 > p) --i;
    const int j = p - ((i * (i + 1)) >> 1);
    const size_t cell = (size_t)(i * 64 + j) * 32;
    const float4 v0 = *(const float4*)(Sb + cell);      // latents l0..l0+3
    const float4 v1 = *(const float4*)(Sb + cell + 4);  // latents l0+4..l0+7
    Ap[0 * 2080 + p] = v0.x;
    Ap[1 * 2080 + p] = v0.y;
    Ap[2 * 2080 + p] = v0.z;
    Ap[3 * 2080 + p] = v0.w;
    Ap[4 * 2080 + p] = v1.x;
    Ap[5 * 2080 + p] = v1.y;
    Ap[6 * 2080 + p] = v1.z;
    Ap[7 * 2080 + p] = v1.w;
  }
  __syncthreads();

  const int m = tid >> 6;  // matrix (latent) index 0..7
  const int r = tid & 63;  // row owned by this thread
  float* A = Ap + m * 2080;
  const int base_r = (r * (r + 1)) >> 1;

  // Trace via symmetry, each thread dots its own row against packed Kinv.
  float val = 0.f;
  {
    const float* kp = KinvP + (size_t)(l0 + m) * 2080 + base_r;
    for (int j = 0; j < r; ++j) val += 2.0f * kp[j] * A[base_r + j];
    val += kp[r] * A[base_r + r];
  }
  __syncthreads();

  // Packed right-looking Cholesky; subtract 2*log(pivot) on the fly.
  for (int k = 0; k < 64; ++k) {
    const int kdiag = ((k * (k + 1)) >> 1) + k;
    if (r == k) {
      float d = sqrtf(A[kdiag]);
      A[kdiag] = d;
      val -= 2.0f * logf(d);
    }
    __syncthreads();
    float c = 0.f;
    if (r > k) {
      c = A[base_r + k] / A[kdiag];
      A[base_r + k] = c;
    }
    __syncthreads();
    if (r > k) {
      int jb = ((k + 1) * (k + 2)) >> 1;  // row base of row j, incremental
      for (int j = k + 1; j <= r; ++j) {
        A[base_r + j] -= c * A[jb + k];
        jb += j + 1;
      }
    }
    __syncthreads();
  }

  // Reduce val (= trace - logdet_S) over the 64 threads of each matrix.
  for (int off = 16; off > 0; off >>= 1) val += __shfl_xor(val, off);
  if ((tid & 31) == 0) wred[tid >> 5] = val;
  __syncthreads();
  if (tid < 8)
    tlds[(size_t)b * 32 + l0 + tid] = wred[2 * tid] + wred[2 * tid + 1];
}

// ---------------------------------------------------------------------------
// Kernel C: maha[b,l] = mu^T K^-1 mu via f32 WMMA. One wave per
// (16-batch tile, latent). Y = Mu(16xT) * Kinv_l(TxT) from
// v_wmma_f32_16x16x4_f32 tiles, then row-dot with Mu and lane reduction.
// ---------------------------------------------------------------------------
__global__ __launch_bounds__(32) void maha_kernel(
    const float* __restrict__ mus, const float* __restrict__ KinvL,
    float* __restrict__ maha) {
  const int bt = blockIdx.x & 15;
  const int l = blockIdx.x >> 4;
  const int b0 = bt * 16;
  const int lane = threadIdx.x;
  const int half = lane >> 4;  // lanes 16-31 hold K+2 / rows M+8
  const int mrow = lane & 15;
  const float* Kl = KinvL + (size_t)l * 4096;

  float mah[8];
#pragma unroll
  for (int v = 0; v < 8; ++v) mah[v] = 0.f;

  for (int n0 = 0; n0 < 64; n0 += 16) {
    v8f c = {};
    for (int kk = 0; kk < 16; ++kk) {
      const int kb = kk * 4 + half * 2;
      v2f a, bv;
      // A (16x4): M = batch row, K = time index   (mus layout [B,T,L])
      a.x = mus[((size_t)(b0 + mrow) * 64 + kb + 0) * 32 + l];
      a.y = mus[((size_t)(b0 + mrow) * 64 + kb + 1) * 32 + l];
      // B (4x16): row K = time, col N = time'     (KinvL layout [L,T,T])
      bv.x = Kl[(kb + 0) * 64 + n0 + mrow];
      bv.y = Kl[(kb + 1) * 64 + n0 + mrow];
      c = __builtin_amdgcn_wmma_f32_16x16x4_f32(false, a, false, bv, (short)0,
                                                c, false, false);
    }
#pragma unroll
    for (int v = 0; v < 8; ++v) {
      const int row = half * 8 + v;  // C/D layout: VGPR v -> M = v + 8*half
      mah[v] += c[v] * mus[((size_t)(b0 + row) * 64 + n0 + mrow) * 32 + l];
    }
  }
#pragma unroll
  for (int v = 0; v < 8; ++v) {  // reduce over the 16 N-lanes of each half
    mah[v] += __shfl_xor(mah[v], 1);
    mah[v] += __shfl_xor(mah[v], 2);
    mah[v] += __shfl_xor(mah[v], 4);
    mah[v] += __shfl_xor(mah[v], 8);
  }
  if (mrow == 0) {
#pragma unroll
    for (int v = 0; v < 8; ++v)
      maha[(size_t)(b0 + half * 8 + v) * 32 + l] = mah[v];
  }
}

// ---------------------------------------------------------------------------
// Kernel D: quad[b] = sum_{s,t,d} (X-m)^2 / R.  One 256-thread block per b;
// float4 streams, X loaded once and reused across the S samples.
// ---------------------------------------------------------------------------
__global__ __launch_bounds__(256) void ll_kernel(
    const float* __restrict__ X, const float* __restrict__ Mn,
    const float* __restrict__ R, float* __restrict__ quad) {
  __shared__ float ir[128];
  __shared__ float red[256];
  const int b = blockIdx.x;
  const int tid = threadIdx.x;
  if (tid < 128) ir[tid] = 1.0f / R[tid];
  __syncthreads();
  float acc = 0.f;
  const float* Xb = X + (size_t)b * 8192;
  const float* Mb = Mn + (size_t)b * 32768;
  for (int q = tid; q < 2048; q += 256) {
    const int f = q * 4;
    const int d = f & 127;
    const float4 x4 = *(const float4*)(Xb + f);
    const float4 r4 = *(const float4*)(ir + d);
#pragma unroll
    for (int s = 0; s < 4; ++s) {
      const float4 m4 = *(const float4*)(Mb + s * 8192 + f);
      const float dx = x4.x - m4.x, dy = x4.y - m4.y;
      const float dz = x4.z - m4.z, dw = x4.w - m4.w;
      acc += dx * dx * r4.x + dy * dy * r4.y + dz * dz * r4.z + dw * dw * r4.w;
    }
  }
  red[tid] = acc;
  __syncthreads();
  for (int s2 = 128; s2 > 0; s2 >>= 1) {
    if (tid < s2) red[tid] += red[tid + s2];
    __syncthreads();
  }
  if (tid == 0) quad[b] = red[0];
}

// ---------------------------------------------------------------------------
// Kernel E: out[b] = 0.5*sum_l(logdetK + (trace-logdetS) + maha - T)
//                    + (0.5/S)*quad[b] + 0.5*T*(sum log R + D*log 2pi)
// ---------------------------------------------------------------------------
__global__ __launch_bounds__(256) void final_kernel(
    const float* __restrict__ R, const float* __restrict__ logdetK,
    const float* __restrict__ tlds, const float* __restrict__ maha,
    const float* __restrict__ quad, float* __restrict__ out) {
  __shared__ float red[256];
  __shared__ float ldk[32];
  const int tid = threadIdx.x;
  red[tid] = (tid < 128) ? logf(R[tid]) : 0.f;
  if (tid < 32) ldk[tid] = logdetK[tid];
  __syncthreads();
  for (int s2 = 128; s2 > 0; s2 >>= 1) {
    if (tid < s2) red[tid] += red[tid + s2];
    __syncthreads();
  }
  const float sumLogR = red[0];
  const float log2pi = 1.8378770664093453f;
  const int b = tid;
  float acc = 0.f;
  for (int l = 0; l < 32; ++l)
    acc += ldk[l] + tlds[b * 32 + l] + maha[b * 32 + l] - 64.0f;
  out[b] = 0.5f * acc + 0.125f * quad[b] + 32.0f * (sumLogR + 128.0f * log2pi);
}

extern "C" void kernel_launch(void* const* d_in, const int* in_sizes, int n_in,
                              void* d_out, int out_size, void* d_ws,
                              size_t ws_size, hipStream_t stream) {
  const float* X = (const float*)d_in[0];    // [256,64,128]
  const float* Mn = (const float*)d_in[1];   // [256,4,64,128]
  const float* R = (const float*)d_in[2];    // [128]
  const float* mus = (const float*)d_in[3];  // [256,64,32]
  const float* Sig = (const float*)d_in[4];  // [256,64,64,32]
  const float* K = (const float*)d_in[5];    // [64,64,32]
  float* out = (float*)d_out;                // [256]

  float* ws = (float*)d_ws;
  float* KinvL = ws;               // 131072 floats  [L,T,T]
  float* KinvP = ws + 131072;      // 66560 floats   [L, T*(T+1)/2] packed
  float* logdetK = ws + 197632;    // 32
  float* tlds = ws + 197664;       // 8192  (trace - logdet_S) per (b,l)
  float* maha = ws + 205856;       // 8192
  float* quad = ws + 214048;       // 256

  prior_prep_kernel<<<32, 64, 0, stream>>>(K, KinvL, KinvP, logdetK);
  sigma_kernel<<<256 * 4, 512, 0, stream>>>(Sig, KinvP, tlds);
  maha_kernel<<<16 * 32, 32, 0, stream>>>(mus, KinvL, maha);
  ll_kernel<<<256, 256, 0, stream>>>(X, Mn, R, quad);
  final_kernel<<<1, 256, 0, stream>>>(R, logdetK, tlds, maha, quad, out);
}